// SelectiveScanCuda_85985245266335
// MI455X (gfx1250) — compile-verified
//
#include <hip/hip_runtime.h>
#include <stdint.h>

#ifndef __has_builtin
#define __has_builtin(x) 0
#endif

#if __has_builtin(__builtin_amdgcn_global_load_async_to_lds_b128)
#define HAVE_ASYNC_LDS 1
#endif
#if __has_builtin(__builtin_amdgcn_s_wait_asynccnt)
#define HAVE_WAIT_ASYNC 1
#endif
#if __has_builtin(__builtin_amdgcn_exp2f)
#define EXP2F(x) __builtin_amdgcn_exp2f(x)
#else
#define EXP2F(x) exp2f(x)
#endif

#define T_CHUNK 64   // timesteps per LDS tile
#define BLOCK   256  // threads per block (8 waves)
#define DTILE   32   // d-channels per block
#define NSTATE  16   // N (fixed by reference)

// Vector-of-4-int type matching the async builtin's pointee type.
typedef int v4i __attribute__((vector_size(16)));
typedef __attribute__((address_space(1))) v4i* gptr_v4i;
typedef __attribute__((address_space(3))) v4i* lptr_v4i;

// 16-byte global -> LDS copy; async (ASYNCcnt) on CDNA5, sync fallback otherwise.
__device__ __forceinline__ void cp16(float* lds_dst, const float* gsrc) {
#if defined(HAVE_ASYNC_LDS)
  __builtin_amdgcn_global_load_async_to_lds_b128(
      (gptr_v4i)gsrc, (lptr_v4i)lds_dst, 0, 0);
#else
  *(float4*)lds_dst = *(const float4*)gsrc;
#endif
}

__device__ __forceinline__ void wait_async_all() {
#if defined(HAVE_ASYNC_LDS)
#if defined(HAVE_WAIT_ASYNC)
  __builtin_amdgcn_s_wait_asynccnt(0);
#else
  asm volatile("s_wait_asynccnt 0" ::: "memory");
#endif
#endif
}

// VALU cross-lane add via DPP16 (no LDS traffic).
// 0xB1 = quad_perm(1,0,3,2)  (xor 1)
// 0x4E = quad_perm(2,3,0,1)  (xor 2)
// 0x141 = ROW_HALF_MIRROR    (pairs quads 0<->1, 2<->3 once quads are uniform)
#define DPP_ADD(v, ctrl)                                                      \
  ((v) + __int_as_float(__builtin_amdgcn_update_dpp(                          \
             0, __float_as_int(v), (ctrl), 0xF, 0xF, true)))

__global__ __launch_bounds__(BLOCK) void selscan_kernel(
    const float* __restrict__ u, const float* __restrict__ delta,
    const float* __restrict__ A, const float* __restrict__ Bs,
    const float* __restrict__ Cs, const float* __restrict__ Ds,
    const float* __restrict__ h_prev, float* __restrict__ out,
    int L, int D) {
  // Double-buffered streaming tiles.
  __shared__ __align__(16) float sD[2][T_CHUNK * DTILE];  // delta [t][d]
  __shared__ __align__(16) float sU[2][T_CHUNK * DTILE];  // u     [t][d]
  __shared__ __align__(16) float sB[2][T_CHUNK * NSTATE]; // B_ssm [t][n]
  __shared__ __align__(16) float sC[2][T_CHUNK * NSTATE]; // C_ssm [t][n]

  const int tid = threadIdx.x;
  const int b   = blockIdx.y;               // 2D grid: no integer division
  const int d0  = blockIdx.x * DTILE;
  const int np  = tid & 7;          // n-pair id (8 per d)
  const int n0  = np * 2;
  const int dl  = tid >> 3;         // 0..31 local d
  const int d   = d0 + dl;

  const size_t bl = (size_t)b * (size_t)L;
  const float* gB = Bs + bl * NSTATE;
  const float* gC = Cs + bl * NSTATE;

  const float kln = 1.44269504088896340736f;  // log2(e): use exp2 path
  const float a0 = A[(size_t)d * NSTATE + n0] * kln;
  const float a1 = A[(size_t)d * NSTATE + n0 + 1] * kln;
  const float dD = Ds[d];
  float h0 = h_prev[((size_t)b * D + d) * NSTATE + n0];
  float h1 = h_prev[((size_t)b * D + d) * NSTATE + n0 + 1];

  auto stage = [&](int c, int buf) {
    const size_t l0 = (size_t)c * T_CHUNK;
    // delta/u tile: T_CHUNK x DTILE floats = 512 16B chunks -> 2 per thread
    int k = tid;
    int t = k >> 3, col = (k & 7) * 4;   // 8 chunks per 32-float row
    cp16(&sD[buf][k * 4], delta + (bl + l0 + t) * D + d0 + col);
    cp16(&sU[buf][k * 4], u     + (bl + l0 + t) * D + d0 + col);
    k = tid + 256;
    t = k >> 3; col = (k & 7) * 4;
    cp16(&sD[buf][k * 4], delta + (bl + l0 + t) * D + d0 + col);
    cp16(&sU[buf][k * 4], u     + (bl + l0 + t) * D + d0 + col);
    // B/C tile: contiguous T_CHUNK*16 floats -> 1 chunk per thread
    cp16(&sB[buf][tid * 4], gB + l0 * NSTATE + tid * 4);
    cp16(&sC[buf][tid * 4], gC + l0 * NSTATE + tid * 4);
  };

  const int nchunks = L / T_CHUNK;
  stage(0, 0);

  float* yp = out + bl * D + d;  // column for this (b, d)

  for (int c = 0; c < nchunks; ++c) {
    const int buf = c & 1;
    wait_async_all();   // this wave's async copies for chunk c have landed
    __syncthreads();    // all waves' copies landed; prev-chunk readers done

    if (c + 1 < nchunks) stage(c + 1, buf ^ 1);  // overlap with compute below

    const float* pD = sD[buf];
    const float* pU = sU[buf];
    const float* pB = sB[buf];
    const float* pC = sC[buf];
    const int lbase = c * T_CHUNK;

#pragma unroll 8
    for (int t = 0; t < T_CHUNK; ++t) {
      const float dlt = pD[t * DTILE + dl];       // LDS broadcast in 8-group
      const float uu  = pU[t * DTILE + dl];
      const float2 bb = *(const float2*)&pB[t * NSTATE + n0];
      const float2 cc = *(const float2*)&pC[t * NSTATE + n0];

      const float e0 = EXP2F(dlt * a0);           // exp(delta*A) via v_exp_f32
      const float e1 = EXP2F(dlt * a1);
      const float x  = dlt * uu;
      h0 = fmaf(e0, h0, x * bb.x);                // h = dA*h + (d*u)*B
      h1 = fmaf(e1, h1, x * bb.y);

      float p = fmaf(h1, cc.y, h0 * cc.x);        // partial sum over 2 n's
      p = DPP_ADD(p, 0xB1);                       // + xor1
      p = DPP_ADD(p, 0x4E);                       // + xor2
      p = DPP_ADD(p, 0x141);                      // + half-mirror -> sum of 16 n
      if (np == 0)
        yp[(size_t)(lbase + t) * D] = fmaf(uu, dD, p);  // y = C.h + u*D
    }
  }
}

extern "C" void kernel_launch(void* const* d_in, const int* in_sizes, int n_in,
                              void* d_out, int out_size, void* d_ws, size_t ws_size,
                              hipStream_t stream) {
  const float* u      = (const float*)d_in[0];
  const float* delta  = (const float*)d_in[1];
  const float* A      = (const float*)d_in[2];
  const float* B_ssm  = (const float*)d_in[3];
  const float* C_ssm  = (const float*)d_in[4];
  const float* D_ssm  = (const float*)d_in[5];
  const float* h_prev = (const float*)d_in[6];
  float* out = (float*)d_out;

  const int D  = in_sizes[5];            // [D]
  const int N  = in_sizes[2] / D;        // [D,N] -> 16
  const int BL = in_sizes[3] / N;        // [B,L,N]
  const int BD = in_sizes[6] / N;        // [B,D,N]
  const int B  = BD / D;
  const int L  = BL / B;

  dim3 grid(D / DTILE, B);
  selscan_kernel<<<grid, BLOCK, 0, stream>>>(u, delta, A, B_ssm, C_ssm, D_ssm,
                                             h_prev, out, L, D);
}